// STAttentionBlock_87591563035155
// MI455X (gfx1250) — compile-verified
//
#include <hip/hip_runtime.h>
#include <hip/hip_bf16.h>

typedef __attribute__((ext_vector_type(2))) float v2f;
typedef __attribute__((ext_vector_type(8))) float v8f;

#define DEVI __device__ __forceinline__

constexpr int N_ = 32, C_ = 64, T_ = 400, V_ = 27, S_ = 2, IC_ = 16;
constexpr int P_ = T_ * V_;                              // 10800 positions / (n,c)
constexpr long long NCP_ = (long long)N_ * C_ * P_;      // 22,118,400
constexpr float SLOPE_ = 0.1f;
constexpr float CNT_INV_ = 1.0f / (float)(N_ * P_);      // BN count = 345600

// padded LDS strides (even, and stride % 64 banks == 2 -> conflict-free A-frags)
constexpr int LW_  = 66;    // 64x64 weights
constexpr int LWO_ = 130;   // 64x128 weight
constexpr int LWT_ = 194;   // 64x192 weight
constexpr int LXS_ = 220;   // 64x216 x-chunk (gcd(220,64)=4 -> conflict-free)

DEVI v8f zero8() { v8f z = {0.f,0.f,0.f,0.f,0.f,0.f,0.f,0.f}; return z; }

// fp32 WMMA 16x16x4 fragment loaders (LDS-resident matrices, row-major)
// A (16x4): lanes 0-15 -> M=lane, K={k,k+1}; lanes 16-31 -> K={k+2,k+3}
DEVI v2f afrag(const float* A, int lda, int m0, int k, int lane) {
  int r  = m0 + (lane & 15);
  int kk = k + ((lane >> 4) << 1);
  v2f a; a.x = A[r * lda + kk]; a.y = A[r * lda + kk + 1]; return a;
}
// B (4x16): lanes give N=lane&15; K pairs as above
DEVI v2f bfrag(const float* B, int ldb, int k, int n0, int lane) {
  int c  = n0 + (lane & 15);
  int kk = k + ((lane >> 4) << 1);
  v2f b; b.x = B[kk * ldb + c]; b.y = B[(kk + 1) * ldb + c]; return b;
}
DEVI v8f wmma4(v2f a, v2f b, v8f acc) {
  return __builtin_amdgcn_wmma_f32_16x16x4_f32(false, a, false, b, (short)0, acc,
                                               false, false);
}
DEVI float leaky(float x) { return x > 0.f ? x : SLOPE_ * x; }

// ---------------------------------------------------------------------------
// zero small workspace regions (graph-replay safe, no hipMemset*)
__global__ void k_zero(float* p, int n) {
  int i = blockIdx.x * 256 + threadIdx.x;
  if (i < n) p[i] = 0.f;
}

// ---------------------------------------------------------------------------
// Kernel 1: qk = W_in @ (x+pe) via WMMA in LDS; q^T k logits via WMMA + atomics
// grid (N, 25), 16 t-steps (432 positions) per block, 256 threads.
__global__ __launch_bounds__(256) void k_att_logits(
    const float* __restrict__ x, const float* __restrict__ pe,
    const float* __restrict__ w_in, const float* __restrict__ b_in,
    float* __restrict__ attL)
{
  extern __shared__ float sm[];
  float* w  = sm;                     // 64 x 66 (padded)
  float* ys = sm + 64 * LW_;          // 64 x 432  (x + pe)
  float* qs = ys + 64 * 432;          // 64 x 432  (qk)
  const int tid = threadIdx.x, lane = tid & 31, wid = tid >> 5;
  const int n = blockIdx.x, t0 = blockIdx.y * 16;
  const long long nbase = (long long)n * C_ * P_;

  // stage weights (pad stride), float4 global loads
  for (int i = tid; i < 64 * 16; i += 256) {
    int o = i >> 4, j = (i & 15) << 2;
    float4 v = ((const float4*)w_in)[i];
    float* d = w + o * LW_ + j;
    d[0] = v.x; d[1] = v.y; d[2] = v.z; d[3] = v.w;
  }
  // stage ys = x + pe (432 = 108 float4 per channel row)
  for (int i = tid; i < 64 * 108; i += 256) {
    int c = i / 108, j = (i - c * 108) << 2;
    long long g = (long long)c * P_ + t0 * V_ + j;
    float4 a = *(const float4*)(x + nbase + g);
    float4 b = *(const float4*)(pe + g);
    float* d = ys + c * 432 + j;
    d[0] = a.x + b.x; d[1] = a.y + b.y; d[2] = a.z + b.z; d[3] = a.w + b.w;
  }
  __syncthreads();

  // qk(64x432) = w(64x64) @ ys(64x432)
  for (int tile = wid; tile < 4 * 27; tile += 8) {
    int m0 = (tile & 3) * 16, n0 = (tile >> 2) * 16;
    v8f acc = zero8();
    for (int k = 0; k < 64; k += 4)
      acc = wmma4(afrag(w, LW_, m0, k, lane), bfrag(ys, 432, k, n0, lane), acc);
    int rbase = m0 + 8 * (lane >> 4);
    int col = n0 + (lane & 15);
    for (int r = 0; r < 8; ++r)
      qs[(rbase + r) * 432 + col] = acc[r] + b_in[rbase + r];
  }
  __syncthreads();

  // partial logits att[s][u][v] += Q^T(27x256) @ K(256x27); K-dim = (cc,tt)=256.
  // 8 tiles (s,2x2 of 16) -> exactly one per wave; uniform EXEC around wmma.
  {
    int tile = wid;
    int s = tile >> 2, m0 = ((tile >> 1) & 1) * 16, n0 = (tile & 1) * 16;
    const float* qbase = qs + (s * 16) * 432;        // q channels
    const float* kbase = qs + ((2 + s) * 16) * 432;  // k channels
    int u = m0 + (lane & 15);                        // A row (may be >=27: masked)
    int vcol = n0 + (lane & 15);                     // B col
    v8f acc = zero8();
    #pragma unroll 4
    for (int k = 0; k < 256; k += 4) {
      int kk = k + ((lane >> 4) << 1);
      int cc0 = kk >> 4, tt0 = kk & 15;
      int kk1 = kk + 1, cc1 = kk1 >> 4, tt1 = kk1 & 15;
      v2f a, b;
      a.x = qbase[cc0 * 432 + tt0 * V_ + u];
      a.y = qbase[cc1 * 432 + tt1 * V_ + u];
      b.x = kbase[cc0 * 432 + tt0 * V_ + vcol];
      b.y = kbase[cc1 * 432 + tt1 * V_ + vcol];
      acc = wmma4(a, b, acc);
    }
    int rbase = m0 + 8 * (lane >> 4);
    if (vcol < V_) {
      for (int r = 0; r < 8; ++r) {
        int uu = rbase + r;
        if (uu < V_)
          atomicAdd(&attL[((n * S_ + s) * V_ + uu) * V_ + vcol], acc[r]);
      }
    }
  }
}

// ---------------------------------------------------------------------------
// Kernel 2: att = tanh(logit / (IC*T)) * alpha + att0   (tiny)
__global__ void k_att_final(const float* __restrict__ attL,
                            const float* __restrict__ alphas,
                            const float* __restrict__ att0,
                            float* __restrict__ att)
{
  int i = blockIdx.x * 256 + threadIdx.x;
  if (i >= N_ * S_ * V_ * V_) return;
  int s  = (i / (V_ * V_)) % S_;
  int uv = i % (S_ * V_ * V_);
  att[i] = tanhf(attL[i] * (1.f / (float)(IC_ * T_))) * alphas[s] + att0[uv];
}

// ---------------------------------------------------------------------------
// Kernel 3: attention apply (rows x 27pad32 @ att) + W_out(64x128) GEMM, fused.
// grid (N, 50), 8 t-steps (216 positions) per block.
__global__ __launch_bounds__(256) void k_apply_wout(
    const float* __restrict__ x, const float* __restrict__ att,
    const float* __restrict__ w_out, const float* __restrict__ b_out,
    float* __restrict__ out1)
{
  extern __shared__ float sm[];
  const int POS = 8 * V_;            // 216
  float* xs = sm;                    // 64 x 220 (padded)
  float* y2 = xs + 64 * LXS_;        // 128 x 216
  float* wo = y2 + 128 * POS;        // 64 x 130 (padded)
  float* ap = wo + 64 * LWO_;        // 2 x 32 x 32, zero-padded att
  const int tid = threadIdx.x, lane = tid & 31, wid = tid >> 5;
  const int n = blockIdx.x, t0 = blockIdx.y * 8;
  const long long nbase = (long long)n * C_ * P_;

  for (int i = tid; i < 64 * 54; i += 256) {            // 216 = 54 float4
    int c = i / 54, j = (i - c * 54) << 2;
    float4 v = *(const float4*)(x + nbase + (long long)c * P_ + t0 * V_ + j);
    float* d = xs + c * LXS_ + j;
    d[0] = v.x; d[1] = v.y; d[2] = v.z; d[3] = v.w;
  }
  for (int i = tid; i < 64 * 32; i += 256) {            // 128 = 32 float4
    int o = i >> 5, j = (i & 31) << 2;
    float4 v = ((const float4*)w_out)[i];
    float* d = wo + o * LWO_ + j;
    d[0] = v.x; d[1] = v.y; d[2] = v.z; d[3] = v.w;
  }
  for (int i = tid; i < 2 * 32 * 32; i += 256) {
    int s = i >> 10, u = (i >> 5) & 31, v = i & 31;
    ap[i] = (u < V_ && v < V_) ? att[((n * S_ + s) * V_ + u) * V_ + v] : 0.f;
  }
  __syncthreads();

  // y2[(s*64+c), t*27+v] = sum_u xs[c][t*27+u] * att[s][u][v], K padded 27->32
  for (int tile = wid; tile < 128; tile += 8) {
    int s = tile >> 6, rest = tile & 63;
    int m0 = (rest >> 1) * 16, n0 = (rest & 1) * 16;   // rows (c,t): 512 -> 32 tiles
    v8f acc = zero8();
    for (int k = 0; k < 32; k += 4) {
      int r = m0 + (lane & 15);
      int c = r >> 3, tt = r & 7;
      int kk = k + ((lane >> 4) << 1);
      v2f a;
      a.x = (kk     < V_) ? xs[c * LXS_ + tt * V_ + kk]     : 0.f;
      a.y = (kk + 1 < V_) ? xs[c * LXS_ + tt * V_ + kk + 1] : 0.f;
      v2f b = bfrag(ap + s * 1024, 32, k, n0, lane);
      acc = wmma4(a, b, acc);
    }
    int rbase = m0 + 8 * (lane >> 4);
    int col = n0 + (lane & 15);
    if (col < V_) {
      for (int r = 0; r < 8; ++r) {
        int row = rbase + r, c = row >> 3, tt = row & 7;
        y2[(s * 64 + c) * POS + tt * V_ + col] = acc[r];
      }
    }
  }
  __syncthreads();

  // out1(64x216) = w_out(64x128) @ y2(128x216); 14 col tiles (tail 8 masked)
  for (int tile = wid; tile < 4 * 14; tile += 8) {
    int m0 = (tile & 3) * 16, n0 = (tile >> 2) * 16;
    v8f acc = zero8();
    for (int k = 0; k < 128; k += 4)
      acc = wmma4(afrag(wo, LWO_, m0, k, lane), bfrag(y2, POS, k, n0, lane), acc);
    int rbase = m0 + 8 * (lane >> 4);
    int col = n0 + (lane & 15);
    if (col < POS) {
      for (int r = 0; r < 8; ++r) {
        int o = rbase + r;
        out1[nbase + (long long)o * P_ + t0 * V_ + col] = acc[r] + b_out[o];
      }
    }
  }
}

// ---------------------------------------------------------------------------
// BN: two-pass, global per-channel stats (reduction + atomics), then finalize.
__global__ __launch_bounds__(256) void bn_stats(const float* __restrict__ buf,
                                                float* __restrict__ sum,
                                                float* __restrict__ sq)
{
  __shared__ float s1[256], s2[256];
  const int tid = threadIdx.x, o = blockIdx.x, n = blockIdx.y;
  const float* p = buf + ((long long)n * C_ + o) * P_;
  float a = 0.f, b = 0.f;
  for (int i = tid; i < P_ / 4; i += 256) {             // 10800 = 2700 float4
    float4 v = ((const float4*)p)[i];
    a += v.x + v.y + v.z + v.w;
    b += v.x * v.x + v.y * v.y + v.z * v.z + v.w * v.w;
  }
  s1[tid] = a; s2[tid] = b; __syncthreads();
  for (int st = 128; st > 0; st >>= 1) {
    if (tid < st) { s1[tid] += s1[tid + st]; s2[tid] += s2[tid + st]; }
    __syncthreads();
  }
  if (tid == 0) { atomicAdd(&sum[o], s1[0]); atomicAdd(&sq[o], s2[0]); }
}

__global__ void bn_finalize(const float* sum, const float* sq,
                            const float* g, const float* b,
                            float* scale, float* shift)
{
  int o = threadIdx.x;
  if (o < C_) {
    float mean = sum[o] * CNT_INV_;
    float var  = sq[o] * CNT_INV_ - mean * mean;
    float inv  = rsqrtf(var + 1e-5f);
    scale[o] = g[o] * inv;
    shift[o] = b[o] - mean * g[o] * inv;
  }
}

// ---------------------------------------------------------------------------
// Kernel 5: y = leaky(x + bn(out1)) in LDS, out2 = W_ff @ y (in-place over buf0)
// grid (N, 25), 432 positions per block.
__global__ __launch_bounds__(256) void k_resid_ff(
    const float* __restrict__ x, float* __restrict__ buf0,
    const float* __restrict__ w_ff, const float* __restrict__ b_ff,
    const float* __restrict__ scale1, const float* __restrict__ shift1)
{
  extern __shared__ float sm[];
  float* ys = sm;                  // 64 x 432
  float* wf = ys + 64 * 432;       // 64 x 66 (padded)
  const int tid = threadIdx.x, lane = tid & 31, wid = tid >> 5;
  const int n = blockIdx.x, t0 = blockIdx.y * 16;
  const long long nbase = (long long)n * C_ * P_;

  for (int i = tid; i < 64 * 16; i += 256) {
    int o = i >> 4, j = (i & 15) << 2;
    float4 v = ((const float4*)w_ff)[i];
    float* d = wf + o * LW_ + j;
    d[0] = v.x; d[1] = v.y; d[2] = v.z; d[3] = v.w;
  }
  for (int i = tid; i < 64 * 108; i += 256) {
    int c = i / 108, j = (i - c * 108) << 2;
    long long g = nbase + (long long)c * P_ + t0 * V_ + j;
    float4 xv = *(const float4*)(x + g);
    float4 bv = *(const float4*)(buf0 + g);
    float s = scale1[c], h = shift1[c];
    float* d = ys + c * 432 + j;
    d[0] = leaky(xv.x + bv.x * s + h);
    d[1] = leaky(xv.y + bv.y * s + h);
    d[2] = leaky(xv.z + bv.z * s + h);
    d[3] = leaky(xv.w + bv.w * s + h);
  }
  __syncthreads();

  for (int tile = wid; tile < 4 * 27; tile += 8) {
    int m0 = (tile & 3) * 16, n0 = (tile >> 2) * 16;
    v8f acc = zero8();
    for (int k = 0; k < 64; k += 4)
      acc = wmma4(afrag(wf, LW_, m0, k, lane), bfrag(ys, 432, k, n0, lane), acc);
    int rbase = m0 + 8 * (lane >> 4);
    int col = n0 + (lane & 15);
    for (int r = 0; r < 8; ++r) {
      int o = rbase + r;
      buf0[nbase + (long long)o * P_ + t0 * V_ + col] = acc[r] + b_ff[o];
    }
  }
}

// ---------------------------------------------------------------------------
// Kernel 7: y3 = leaky(x + bn(out2))   (elementwise streaming)
__global__ void k_resid2(const float* __restrict__ x, const float* __restrict__ buf0,
                         const float* __restrict__ scale2, const float* __restrict__ shift2,
                         float* __restrict__ y3)
{
  long long g = (long long)blockIdx.x * 256 + threadIdx.x;
  if (g >= NCP_) return;
  if ((g & 63) == 0) {                       // gfx1250 global_prefetch_b8
    __builtin_prefetch(x + g + 16384, 0, 0);
    __builtin_prefetch(buf0 + g + 16384, 0, 0);
  }
  int c = (int)((g / P_) % C_);
  y3[g] = leaky(x[g] + buf0[g] * scale2[c] + shift2[c]);
}

// ---------------------------------------------------------------------------
// Kernel 8: temporal conv (3 taps) as im2col GEMM (64x192) @ (192x432) w/ halo
// grid (N, 25).
__global__ __launch_bounds__(256) void k_tconv(
    const float* __restrict__ y3, const float* __restrict__ w_t,
    const float* __restrict__ b_t, float* __restrict__ z)
{
  extern __shared__ float sm[];
  const int HQ = 18 * V_;            // 486: 16 t-steps + 2 halo
  float* yh = sm;                    // 64 x 486
  float* wt = yh + 64 * HQ;          // 64 x 194 (padded), [o][dt*64+i]
  const int tid = threadIdx.x, lane = tid & 31, wid = tid >> 5;
  const int n = blockIdx.x, t0 = blockIdx.y * 16;
  const long long nbase = (long long)n * C_ * P_;

  for (int i = tid; i < 64 * 192; i += 256) {
    int o = i / 192, k = i - o * 192;
    int dt = k >> 6, ii = k & 63;
    wt[o * LWT_ + k] = w_t[(o * 64 + ii) * 3 + dt];
  }
  for (int i = tid; i < 64 * HQ; i += 256) {
    int c = i / HQ, q = i - c * HQ;
    int gt = t0 - 1 + q / V_, vv = q % V_;
    yh[c * HQ + q] = (gt >= 0 && gt < T_)
        ? y3[nbase + (long long)c * P_ + gt * V_ + vv] : 0.f;
  }
  __syncthreads();

  for (int tile = wid; tile < 4 * 27; tile += 8) {
    int m0 = (tile & 3) * 16, n0 = (tile >> 2) * 16;
    int col = n0 + (lane & 15);
    v8f acc = zero8();
    for (int k = 0; k < 192; k += 4) {
      v2f a = afrag(wt, LWT_, m0, k, lane);
      int kk = k + ((lane >> 4) << 1), kk2 = kk + 1;
      v2f b;
      b.x = yh[(kk  & 63) * HQ + col + (kk  >> 6) * V_];  // B[k][p] = y3h[i][p+dt*27]
      b.y = yh[(kk2 & 63) * HQ + col + (kk2 >> 6) * V_];
      acc = wmma4(a, b, acc);
    }
    int rbase = m0 + 8 * (lane >> 4);
    for (int r = 0; r < 8; ++r) {
      int o = rbase + r;
      z[nbase + (long long)o * P_ + t0 * V_ + col] = acc[r] + b_t[o];
    }
  }
}

// ---------------------------------------------------------------------------
// Kernel 10: out = leaky(y3 + bn(z))
__global__ void k_final(const float* __restrict__ y3, const float* __restrict__ z,
                        const float* __restrict__ scale3, const float* __restrict__ shift3,
                        float* __restrict__ out)
{
  long long g = (long long)blockIdx.x * 256 + threadIdx.x;
  if (g >= NCP_) return;
  if ((g & 63) == 0) {
    __builtin_prefetch(y3 + g + 16384, 0, 0);
    __builtin_prefetch(z + g + 16384, 0, 0);
  }
  int c = (int)((g / P_) % C_);
  out[g] = leaky(y3[g] + z[g] * scale3[c] + shift3[c]);
}

// ---------------------------------------------------------------------------
extern "C" void kernel_launch(void* const* d_in, const int* in_sizes, int n_in,
                              void* d_out, int out_size, void* d_ws, size_t ws_size,
                              hipStream_t stream)
{
  const float* x      = (const float*)d_in[0];
  const float* pe     = (const float*)d_in[1];
  const float* w_in   = (const float*)d_in[2];
  const float* b_in   = (const float*)d_in[3];
  const float* alphas = (const float*)d_in[4];
  const float* att0   = (const float*)d_in[5];
  const float* w_out  = (const float*)d_in[6];
  const float* b_out  = (const float*)d_in[7];
  const float* g_out  = (const float*)d_in[8];
  const float* be_out = (const float*)d_in[9];
  const float* w_ff   = (const float*)d_in[10];
  const float* b_ff   = (const float*)d_in[11];
  const float* g_ff   = (const float*)d_in[12];
  const float* be_ff  = (const float*)d_in[13];
  const float* w_t    = (const float*)d_in[14];
  const float* b_t    = (const float*)d_in[15];
  const float* g_t    = (const float*)d_in[16];
  const float* be_t   = (const float*)d_in[17];
  float* out = (float*)d_out;

  // workspace layout (floats): buf0 | buf2 | attL | sums(6x64) | att | sscl(6x64)
  float* ws   = (float*)d_ws;
  float* buf0 = ws;                       // out1 -> out2 (in place) -> z
  float* buf2 = ws + NCP_;                // y3
  float* attL = ws + 2 * NCP_;            // 32*2*27*27 = 46656
  float* sums = attL + N_ * S_ * V_ * V_; // 6*64
  float* att  = sums + 6 * 64;            // 46656
  float* sscl = att + N_ * S_ * V_ * V_;  // 6*64

  const int ATTN = N_ * S_ * V_ * V_;
  int zn = ATTN + 6 * 64;                 // attL + sums are contiguous
  k_zero<<<(zn + 255) / 256, 256, 0, stream>>>(attL, zn);

  k_att_logits<<<dim3(N_, 25), 256,
      (64 * LW_ + 2 * 64 * 432) * sizeof(float), stream>>>(
      x, pe, w_in, b_in, attL);
  k_att_final<<<(ATTN + 255) / 256, 256, 0, stream>>>(attL, alphas, att0, att);

  k_apply_wout<<<dim3(N_, 50), 256,
      (64 * LXS_ + 128 * 216 + 64 * LWO_ + 2 * 32 * 32) * sizeof(float), stream>>>(
      x, att, w_out, b_out, buf0);
  bn_stats<<<dim3(C_, N_), 256, 0, stream>>>(buf0, sums + 0, sums + 64);
  bn_finalize<<<1, 64, 0, stream>>>(sums + 0, sums + 64, g_out, be_out,
                                    sscl + 0, sscl + 64);

  k_resid_ff<<<dim3(N_, 25), 256,
      (64 * 432 + 64 * LW_) * sizeof(float), stream>>>(
      x, buf0, w_ff, b_ff, sscl + 0, sscl + 64);
  bn_stats<<<dim3(C_, N_), 256, 0, stream>>>(buf0, sums + 128, sums + 192);
  bn_finalize<<<1, 64, 0, stream>>>(sums + 128, sums + 192, g_ff, be_ff,
                                    sscl + 128, sscl + 192);

  k_resid2<<<(int)((NCP_ + 255) / 256), 256, 0, stream>>>(x, buf0, sscl + 128,
                                                          sscl + 192, buf2);

  k_tconv<<<dim3(N_, 25), 256,
      (64 * 486 + 64 * LWT_) * sizeof(float), stream>>>(
      buf2, w_t, b_t, buf0);
  bn_stats<<<dim3(C_, N_), 256, 0, stream>>>(buf0, sums + 256, sums + 320);
  bn_finalize<<<1, 64, 0, stream>>>(sums + 256, sums + 320, g_t, be_t,
                                    sscl + 256, sscl + 320);

  k_final<<<(int)((NCP_ + 255) / 256), 256, 0, stream>>>(buf2, buf0, sscl + 256,
                                                         sscl + 320, out);

  (void)in_sizes; (void)n_in; (void)out_size; (void)ws_size;
}